// Detection_78477642432928
// MI455X (gfx1250) — compile-verified
//
#include <hip/hip_runtime.h>
#include <stdint.h>

// Problem geometry (fixed by the reference)
#define G_      52
#define GG      2704          // 52*52
#define CH      85            // 5 + 80 classes
#define NA      3             // anchors
#define CIN     255           // 3*85
#define TILE    128           // spatial positions per TDM tile
#define LDS_STR 129           // 128 + 1 dword pad per row (bank-conflict-free)
#define NT      22            // ceil(2704/128)
#define STRIDE_F 8.0f         // 416/52

typedef unsigned int u32x4 __attribute__((ext_vector_type(4)));
typedef int          i32x8 __attribute__((ext_vector_type(8)));
typedef int          i32x4 __attribute__((ext_vector_type(4)));

#if defined(__has_builtin)
#if __has_builtin(__builtin_amdgcn_tensor_load_to_lds)
#define HAVE_TDM 1
#endif
#endif

__device__ __forceinline__ float sigf(float v) {
    return 1.0f / (1.0f + __expf(-v));
}

__global__ __launch_bounds__(256)
void yolo_decode_kernel(const float* __restrict__ x, float* __restrict__ out) {
    // 85 rows x (128 data + 1 pad) dwords
    __shared__ float tile[CH * LDS_STR];

    const int bid = blockIdx.x;
    const int t   = bid % NT;
    const int a   = (bid / NT) % NA;
    const int b   = bid / (NT * NA);
    const int s0  = t * TILE;
    const int rem = GG - s0;              // valid spatial positions in this tile

    // start of (b, a, channel 0) row at spatial offset s0
    const size_t inBase = ((size_t)b * CIN + (size_t)a * CH) * GG + (size_t)s0;

#if HAVE_TDM
    // One wave issues the tensor DMA: 2D tile, 85 rows x 128 cols of f32,
    // row stride 2704 floats, LDS padded 1 dword per 128 dwords.
    if (threadIdx.x < 32) {
        const uint64_t gaddr = (uint64_t)(uintptr_t)(x + inBase);
        const uint32_t laddr = (uint32_t)(uintptr_t)(&tile[0]); // LDS byte offset (addr[31:0])

        u32x4 g0;
        g0[0] = 1u;                                   // count = 1 (valid descriptor)
        g0[1] = laddr;                                // lds_addr (bytes)
        g0[2] = (uint32_t)gaddr;                      // global_addr[31:0]
        g0[3] = (uint32_t)((gaddr >> 32) & 0x01FFFFFFu) // global_addr[56:32]
              | (2u << 30);                           // type = 2 ("image")

        i32x8 g1;
        g1[0] = (int)((2u << 16)      // data_size = 2 -> 4 bytes
                    | (1u << 20)      // pad_enable
                    | (6u << 22));    // pad_interval = 6 -> every 128 dwords; pad_amount=0 -> 1 dword
        g1[1] = (int)(((uint32_t)rem & 0xFFFFu) << 16); // tensor_dim0[15:0] = remaining cols (OOB -> 0)
        g1[2] = (int)((uint32_t)CH << 16);              // tensor_dim0 hi = 0 | tensor_dim1[15:0] = 85
        g1[3] = (int)((uint32_t)TILE << 16);            // tensor_dim1 hi = 0 | tile_dim0 = 128
        g1[4] = CH;                                     // tile_dim1 = 85 | tile_dim2 = 0
        g1[5] = GG;                                     // tensor_dim0_stride[31:0] = 2704
        g1[6] = 0;                                      // stride hi | tensor_dim1_stride lo
        g1[7] = 0;

        i32x4 gz4 = {0, 0, 0, 0};                       // groups 2/3 unused (2D tensor)
        i32x8 gz8 = {0, 0, 0, 0, 0, 0, 0, 0};
        __builtin_amdgcn_tensor_load_to_lds(g0, g1, gz4, gz4, gz8, 0);
        __builtin_amdgcn_s_wait_tensorcnt(0);           // s_wait_tensorcnt 0
    }
    __syncthreads();
#else
    // Fallback: manual coalesced staging (128-float contiguous runs per channel row)
    for (int idx = threadIdx.x; idx < CH * TILE; idx += 256) {
        const int c = idx >> 7;
        const int s = idx & (TILE - 1);
        tile[c * LDS_STR + s] = (s < rem) ? x[inBase + (size_t)c * GG + s] : 0.0f;
    }
    __syncthreads();
#endif

    // anchor * stride / stride == raw anchor (stride = 8)
    const float AW = (a == 0) ? 10.0f : (a == 1) ? 16.0f : 33.0f;
    const float AH = (a == 0) ? 13.0f : (a == 1) ? 30.0f : 23.0f;

    // Output: contiguous [spatial, channel] span for this tile
    const size_t outBase = (((size_t)b * NA + (size_t)a) * GG + (size_t)s0) * CH;

    const int nElem = TILE * CH; // 10880
    for (int e = threadIdx.x; e < nElem; e += 256) {
        const int s = e / CH;            // spatial within tile
        const int c = e - s * CH;        // channel
        if (s < rem) {
            const float v = tile[c * LDS_STR + s];  // stride 129 -> conflict-free
            float r;
            if (c >= 4) {
                r = sigf(v);                         // conf + classes
            } else if (c == 0) {
                const int gx = (s0 + s) % G_;
                r = (sigf(v) + (float)gx) * STRIDE_F;
            } else if (c == 1) {
                const int gy = (s0 + s) / G_;
                r = (sigf(v) + (float)gy) * STRIDE_F;
            } else if (c == 2) {
                r = __expf(v) * AW;
            } else {
                r = __expf(v) * AH;
            }
            out[outBase + (size_t)s * CH + c] = r;   // fully coalesced b32 stores
        }
    }
}

extern "C" void kernel_launch(void* const* d_in, const int* in_sizes, int n_in,
                              void* d_out, int out_size, void* d_ws, size_t ws_size,
                              hipStream_t stream) {
    const float* x  = (const float*)d_in[0];
    float*       out = (float*)d_out;

    const int B = in_sizes[0] / (CIN * GG);     // 64 for the reference setup
    const int nBlocks = B * NA * NT;

    hipLaunchKernelGGL(yolo_decode_kernel, dim3(nBlocks), dim3(256), 0, stream, x, out);
}